// Grid4DSpatiotemporalEncoder_48610439856780
// MI455X (gfx1250) — compile-verified
//
#include <hip/hip_runtime.h>
#include <math.h>

// ---------------------------------------------------------------------------
// Grid4D spatiotemporal hash-grid encoder for MI455X (gfx1250, wave32).
//
// Per point: 4 tables x 16 levels x 8 gathers (float2) + trilinear FMAs.
// Memory-system bound: keep 64 x 4MB level tables hot in the 192MB L2 by
// streaming the 512MB output with TH=NT stores (CDNA5 temporal hints).
// No WMMA: the 1x8 * 8x2 per-point contraction has a per-point B operand,
// which cannot use the wave-shared-matrix WMMA model profitably.
// ---------------------------------------------------------------------------

typedef float v4f __attribute__((ext_vector_type(4)));

#define TBL_SIZE   (1u << 19)
#define HASH_MASK  (TBL_SIZE - 1u)
#define PRIME1     2654435761u
#define PRIME2     805459861u
#define LVL        16

struct ResPack {
    float sp[LVL];     // spatial resf (all 3 dims equal: base 16 -> max 512)
    float tmxy[LVL];   // temporal resf dims 0,1 (base 8 -> max 32)
    float tmz[LVL];    // temporal resf dim 2   (base 8 -> max 16)
};

// One hash-grid level: trilinear interpolation of 8 gathered float2 corners.
__device__ __forceinline__ void enc_level(const float2* __restrict__ tab,
                                          float ux, float uy, float uz,
                                          float rx, float ry, float rz,
                                          float& f0, float& f1) {
    const float rx1 = rx - 1.0f, ry1 = ry - 1.0f, rz1 = rz - 1.0f;
    // pos = clip(u*(res-1), 0, res-1)
    const float px = fminf(fmaxf(ux * rx1, 0.0f), rx1);
    const float py = fminf(fmaxf(uy * ry1, 0.0f), ry1);
    const float pz = fminf(fmaxf(uz * rz1, 0.0f), rz1);
    const float fx = floorf(px), fy = floorf(py), fz = floorf(pz);
    const float wx1 = px - fx, wy1 = py - fy, wz1 = pz - fz;
    const float wx0 = 1.0f - wx1, wy0 = 1.0f - wy1, wz0 = 1.0f - wz1;
    const unsigned cx = (unsigned)fx, cy = (unsigned)fy, cz = (unsigned)fz;
    // hash terms: idx = (x*1 ^ y*P1 ^ z*P2) & (T-1); share partial products
    const unsigned hx0 = cx,          hx1 = cx + 1u;
    const unsigned hy0 = cy * PRIME1, hy1 = hy0 + PRIME1;
    const unsigned hz0 = cz * PRIME2, hz1 = hz0 + PRIME2;
    // 8 independent b64 gathers -> deep LOADcnt pipelining per level
    const float2 v000 = tab[(hx0 ^ hy0 ^ hz0) & HASH_MASK];
    const float2 v100 = tab[(hx1 ^ hy0 ^ hz0) & HASH_MASK];
    const float2 v010 = tab[(hx0 ^ hy1 ^ hz0) & HASH_MASK];
    const float2 v110 = tab[(hx1 ^ hy1 ^ hz0) & HASH_MASK];
    const float2 v001 = tab[(hx0 ^ hy0 ^ hz1) & HASH_MASK];
    const float2 v101 = tab[(hx1 ^ hy0 ^ hz1) & HASH_MASK];
    const float2 v011 = tab[(hx0 ^ hy1 ^ hz1) & HASH_MASK];
    const float2 v111 = tab[(hx1 ^ hy1 ^ hz1) & HASH_MASK];
    // trilinear weights: offs[c][d] = (c>>d)&1 -> corner (b2,b1,b0)=(z,y,x)
    const float wq00 = wy0 * wz0, wq10 = wy1 * wz0;
    const float wq01 = wy0 * wz1, wq11 = wy1 * wz1;
    const float w000 = wx0 * wq00, w100 = wx1 * wq00;
    const float w010 = wx0 * wq10, w110 = wx1 * wq10;
    const float w001 = wx0 * wq01, w101 = wx1 * wq01;
    const float w011 = wx0 * wq11, w111 = wx1 * wq11;
    f0 = fmaf(w000, v000.x, fmaf(w100, v100.x, fmaf(w010, v010.x,
         fmaf(w110, v110.x, fmaf(w001, v001.x, fmaf(w101, v101.x,
         fmaf(w011, v011.x, w111 * v111.x)))))));
    f1 = fmaf(w000, v000.y, fmaf(w100, v100.y, fmaf(w010, v010.y,
         fmaf(w110, v110.y, fmaf(w001, v001.y, fmaf(w101, v101.y,
         fmaf(w011, v011.y, w111 * v111.y)))))));
}

// Stream a 32-float row (one or more full 128B cache lines) with NT stores so
// the output never evicts the hash tables from L2.
__device__ __forceinline__ void store_row_nt(float* __restrict__ dst,
                                             const float* __restrict__ acc) {
    v4f* d = (v4f*)dst;
#pragma unroll
    for (int j = 0; j < 8; ++j) {
        v4f v = {acc[4 * j + 0], acc[4 * j + 1], acc[4 * j + 2], acc[4 * j + 3]};
        __builtin_nontemporal_store(v, d + j);
    }
}

__global__ __launch_bounds__(256) void grid4d_encode_kernel(
    const float* __restrict__ xyzt,
    const float2* __restrict__ xyz_tab, const float2* __restrict__ xyt_tab,
    const float2* __restrict__ yzt_tab, const float2* __restrict__ xzt_tab,
    float* __restrict__ out, int N, ResPack R) {
    const int i = blockIdx.x * blockDim.x + threadIdx.x;
    if (i >= N) return;

    // one-shot coalesced input read; NT so it doesn't occupy L2 either
    const v4f p = __builtin_nontemporal_load((const v4f*)xyzt + i);
    const float x = p.x, y = p.y, z = p.z;
    const float t = (p.w * 2.0f - 1.0f) * 0.9f;  // exact op order of reference

    // u = (coord + bound) / (2*bound), bound == 1 for all four encodings
    const float ux = (x + 1.0f) * 0.5f;
    const float uy = (y + 1.0f) * 0.5f;
    const float uz = (z + 1.0f) * 0.5f;
    const float ut = (t + 1.0f) * 0.5f;

    float acc[32];

    // ---- spatial: xyz table, cubic resolutions --------------------------
#pragma unroll
    for (int l = 0; l < LVL; ++l) {
        const float r = R.sp[l];
        enc_level(xyz_tab + (size_t)l * TBL_SIZE, ux, uy, uz, r, r, r,
                  acc[2 * l], acc[2 * l + 1]);
    }
    store_row_nt(out + (size_t)i * 32, acc);

    float* tbase = out + (size_t)N * 32 + (size_t)i * 96;

    // ---- temporal: xyt --------------------------------------------------
#pragma unroll
    for (int l = 0; l < LVL; ++l) {
        enc_level(xyt_tab + (size_t)l * TBL_SIZE, ux, uy, ut,
                  R.tmxy[l], R.tmxy[l], R.tmz[l], acc[2 * l], acc[2 * l + 1]);
    }
    store_row_nt(tbase + 0, acc);

    // ---- temporal: yzt --------------------------------------------------
#pragma unroll
    for (int l = 0; l < LVL; ++l) {
        enc_level(yzt_tab + (size_t)l * TBL_SIZE, uy, uz, ut,
                  R.tmxy[l], R.tmxy[l], R.tmz[l], acc[2 * l], acc[2 * l + 1]);
    }
    store_row_nt(tbase + 32, acc);

    // ---- temporal: xzt --------------------------------------------------
#pragma unroll
    for (int l = 0; l < LVL; ++l) {
        enc_level(xzt_tab + (size_t)l * TBL_SIZE, ux, uz, ut,
                  R.tmxy[l], R.tmxy[l], R.tmz[l], acc[2 * l], acc[2 * l + 1]);
    }
    store_row_nt(tbase + 64, acc);
}

extern "C" void kernel_launch(void* const* d_in, const int* in_sizes, int n_in,
                              void* d_out, int out_size, void* d_ws, size_t ws_size,
                              hipStream_t stream) {
    (void)n_in; (void)out_size; (void)d_ws; (void)ws_size;
    const float* xyzt = (const float*)d_in[0];
    const float2* xyz_tab = (const float2*)d_in[1];
    const float2* xyt_tab = (const float2*)d_in[2];
    const float2* yzt_tab = (const float2*)d_in[3];
    const float2* xzt_tab = (const float2*)d_in[4];
    float* out = (float*)d_out;
    const int N = in_sizes[0] / 4;

    // Replicate numpy's _level_resolutions bit-for-bit (double libm ops):
    //   scale = exp2(log2(max/base)/(L-1)); res_l = ceil(base * scale**l)
    ResPack R;
    {
        const double s_sp = exp2(log2(512.0 / 16.0) / 15.0);   // spatial, all dims
        const double s_t01 = exp2(log2(32.0 / 8.0) / 15.0);    // temporal dims 0,1
        const double s_t2 = exp2(log2(16.0 / 8.0) / 15.0);     // temporal dim 2
        for (int l = 0; l < LVL; ++l) {
            R.sp[l]   = (float)(long long)ceil(16.0 * pow(s_sp,  (double)l));
            R.tmxy[l] = (float)(long long)ceil(8.0  * pow(s_t01, (double)l));
            R.tmz[l]  = (float)(long long)ceil(8.0  * pow(s_t2,  (double)l));
        }
    }

    const int block = 256;
    const int grid = (N + block - 1) / block;
    hipLaunchKernelGGL(grid4d_encode_kernel, dim3(grid), dim3(block), 0, stream,
                       xyzt, xyz_tab, xyt_tab, yzt_tab, xzt_tab, out, N, R);
}